// LSTCWA_61469571940555
// MI455X (gfx1250) — compile-verified
//
#include <hip/hip_runtime.h>
#include <math.h>

// Problem constants
#define NROWS   32768
#define DDIM    512
#define LTOK    64
#define SEGLEN  512     // NROWS / LTOK
#define NWIN    16      // ceil(SEGLEN / 32)
#define WINSZ   64
#define STRIDE  32

typedef float v2f __attribute__((ext_vector_type(2)));
typedef float v8f __attribute__((ext_vector_type(8)));

// ---------------------------------------------------------------------------
// Coord mean / std (ddof=1) over N=32768 points, single block.
// ---------------------------------------------------------------------------
__global__ void coord_stats_kernel(const float* __restrict__ coords, float* __restrict__ cst) {
    __shared__ float sx[256], sy[256];
    int tid = threadIdx.x;
    float ax = 0.f, ay = 0.f;
    for (int i = tid; i < NROWS; i += 256) { ax += coords[2 * i]; ay += coords[2 * i + 1]; }
    sx[tid] = ax; sy[tid] = ay; __syncthreads();
    for (int s = 128; s > 0; s >>= 1) {
        if (tid < s) { sx[tid] += sx[tid + s]; sy[tid] += sy[tid + s]; }
        __syncthreads();
    }
    float mx = sx[0] * (1.f / NROWS), my = sy[0] * (1.f / NROWS);
    __syncthreads();
    ax = 0.f; ay = 0.f;
    for (int i = tid; i < NROWS; i += 256) {
        float dx = coords[2 * i] - mx, dy = coords[2 * i + 1] - my;
        ax += dx * dx; ay += dy * dy;
    }
    sx[tid] = ax; sy[tid] = ay; __syncthreads();
    for (int s = 128; s > 0; s >>= 1) {
        if (tid < s) { sx[tid] += sx[tid + s]; sy[tid] += sy[tid + s]; }
        __syncthreads();
    }
    if (tid == 0) {
        cst[0] = mx; cst[1] = my;
        cst[2] = sqrtf(sx[0] / (float)(NROWS - 1)) + 1e-8f;
        cst[3] = sqrtf(sy[0] / (float)(NROWS - 1)) + 1e-8f;
    }
}

// ---------------------------------------------------------------------------
// Zero scratch
// ---------------------------------------------------------------------------
__global__ void zero_kernel(float* __restrict__ p, int n) {
    int i = blockIdx.x * 256 + threadIdx.x;
    if (i < n) p[i] = 0.f;
}

// ---------------------------------------------------------------------------
// Small fp32 GEMM with V_WMMA_F32_16X16X4_F32, templated on transB/bias so
// the inner loop is branch-free: 8 operand loads then 4 back-to-back WMMAs.
//   C[M,N] = A[M,K] @ op(B) (+bias).  TRANSB: B stored [N,K] (use B^T).
// One wave (32 threads) per 16x16 C tile.
// A 16x4 per step: lane(0..15) m=lane, VGPR0=K0,VGPR1=K1 ; lanes 16..31 K2,K3.
// B 4x16 mirrored: lane holds column n=lane&15, K pair selected by lane>>4.
// C: VGPR j -> row = j + 8*(lane>>4), col = lane&15.  K must be %16.
// ---------------------------------------------------------------------------
template <bool TRANSB, bool HASBIAS>
__global__ __launch_bounds__(32) void gemm_wmma_f32(
        const float* __restrict__ A, const float* __restrict__ B,
        const float* __restrict__ bias, float* __restrict__ C,
        int M, int N, int K) {
    int lane  = threadIdx.x & 31;
    int hl    = lane >> 4;       // 0 or 1: selects K pair
    int i16   = lane & 15;
    int ntile = N >> 4;
    int tm    = (blockIdx.x / ntile) << 4;
    int tn    = (blockIdx.x % ntile) << 4;

    const float* arow = A + (size_t)(tm + i16) * K;      // A row for this lane
    const float* bt   = TRANSB ? (B + (size_t)(tn + i16) * K) : (B + tn + i16);

    v8f acc = {};
    for (int k = 0; k < K; k += 16) {
        v2f a[4], b[4];
#pragma unroll
        for (int u = 0; u < 4; ++u) {
            int ka = k + 4 * u + 2 * hl;
            a[u] = *(const v2f*)(arow + ka);
            if (TRANSB) {
                b[u] = *(const v2f*)(bt + ka);
            } else {
                b[u].x = bt[(size_t)ka * N];
                b[u].y = bt[(size_t)(ka + 1) * N];
            }
        }
#pragma unroll
        for (int u = 0; u < 4; ++u) {
            acc = __builtin_amdgcn_wmma_f32_16x16x4_f32(
                /*neg_a=*/false, a[u], /*neg_b=*/false, b[u],
                /*c_mod=*/(short)0, acc, /*reuse_a=*/false, /*reuse_b=*/false);
        }
    }
#pragma unroll
    for (int j = 0; j < 8; ++j) {
        int row = tm + j + 8 * hl;
        int col = tn + i16;
        float v = acc[j];
        if (HASBIAS) v += bias[col];
        C[(size_t)row * N + col] = v;
    }
}

// ---------------------------------------------------------------------------
// Per-row LayerNorm stats + GEMV:  dot[row] = qk[l] . LN(feats[row])
// One wave per row; lane-contiguous 16-float chunks -> global_load_b128 x4.
// ---------------------------------------------------------------------------
__global__ void dot_kernel(const float* __restrict__ feats, const float* __restrict__ qk,
                           float* __restrict__ dotv, float* __restrict__ rmean,
                           float* __restrict__ rrstd) {
    int row  = blockIdx.x * 8 + (threadIdx.x >> 5);
    int lane = threadIdx.x & 31;
    const float* fr = feats + (size_t)row * DDIM + lane * 16;
    const float* qr = qk + (size_t)(row >> 9) * DDIM + lane * 16;

    float fv[16];
    float s = 0.f, sq = 0.f;
#pragma unroll
    for (int t = 0; t < 16; ++t) {
        float x = fr[t];
        fv[t] = x; s += x; sq += x * x;
    }
#pragma unroll
    for (int off = 16; off > 0; off >>= 1) {
        s  += __shfl_xor(s, off);
        sq += __shfl_xor(sq, off);
    }
    float mean = s * (1.f / DDIM);
    float var  = sq * (1.f / DDIM) - mean * mean;
    float rstd = rsqrtf(var + 1e-5f);

    float dfq = 0.f, qs = 0.f;
#pragma unroll
    for (int t = 0; t < 16; ++t) {
        float qv = qr[t];
        dfq += fv[t] * qv;
        qs  += qv;
    }
#pragma unroll
    for (int off = 16; off > 0; off >>= 1) {
        dfq += __shfl_xor(dfq, off);
        qs  += __shfl_xor(qs, off);
    }
    if (lane == 0) {
        dotv[row]  = rstd * (dfq - mean * qs);
        rmean[row] = mean;
        rrstd[row] = rstd;
    }
}

// ---------------------------------------------------------------------------
// Block reductions over 64 threads (2 waves) via LDS.
// ---------------------------------------------------------------------------
__device__ __forceinline__ float bsum64(float v, float* red) {
    int tid = threadIdx.x;
    __syncthreads();
    red[tid] = v;
    __syncthreads();
    for (int s = 32; s > 0; s >>= 1) {
        if (tid < s) red[tid] += red[tid + s];
        __syncthreads();
    }
    return red[0];
}
__device__ __forceinline__ float bmax64(float v, float* red) {
    int tid = threadIdx.x;
    __syncthreads();
    red[tid] = v;
    __syncthreads();
    for (int s = 32; s > 0; s >>= 1) {
        if (tid < s) red[tid] = fmaxf(red[tid], red[tid + s]);
        __syncthreads();
    }
    return red[0];
}

// ---------------------------------------------------------------------------
// Per-window attention: centroid, pb-logit term via (q@pos_w2).h, clip,
// masked softmax, scatter attention weights (divided by nw) to weight[row].
// One block (64 threads) per (l, n) window.
// ---------------------------------------------------------------------------
__global__ void attn_kernel(const float* __restrict__ coords, const float* __restrict__ cst,
                            const float* __restrict__ q, const float* __restrict__ qw2,
                            const float* __restrict__ pw1, const float* __restrict__ pb1,
                            const float* __restrict__ pb2, const float* __restrict__ dotv,
                            float* __restrict__ weight) {
    __shared__ float s_qw2[DDIM], s_w1x[DDIM], s_w1y[DDIM], s_b1[DDIM];
    __shared__ float red[64];

    int tid = threadIdx.x;               // 0..63 == w
    int bid = blockIdx.x;
    int l   = bid >> 4;
    int n   = bid & 15;

    // Preload window-invariant vectors; accumulate q[l].pos_b2 on the side.
    float qbp = 0.f;
    for (int e = tid; e < DDIM; e += 64) {
        s_qw2[e] = qw2[(size_t)l * DDIM + e];
        s_w1x[e] = pw1[2 * e];
        s_w1y[e] = pw1[2 * e + 1];
        s_b1[e]  = pb1[e];
        qbp += q[(size_t)l * DDIM + e] * pb2[e];
    }
    float qb2 = bsum64(qbp, red);

    int  pos   = n * STRIDE + tid;
    bool valid = pos < SEGLEN;
    int  pcm   = valid ? pos : (SEGLEN - 1);
    int  row   = l * SEGLEN + pcm;

    float cx = (coords[2 * row]     - cst[0]) / cst[2];
    float cy = (coords[2 * row + 1] - cst[1]) / cst[3];

    float cnt = (float)((SEGLEN - n * STRIDE) < WINSZ ? (SEGLEN - n * STRIDE) : WINSZ);
    float sx  = bsum64(valid ? cx : 0.f, red);
    float sy  = bsum64(valid ? cy : 0.f, red);
    float dx  = cx - sx / cnt;
    float dy  = cy - sy / cnt;

    // ph = sum_e qw2[e] * relu(w1x[e]*dx + w1y[e]*dy + b1[e])
    float ph = 0.f;
    for (int e = 0; e < DDIM; ++e)
        ph += s_qw2[e] * fmaxf(fmaf(s_w1x[e], dx, fmaf(s_w1y[e], dy, s_b1[e])), 0.f);

    const float invtemp = 0.044194173824159216f; // 1/sqrt(512)
    float logit = (dotv[row] + ph + qb2) * invtemp;
    logit = fminf(fmaxf(logit, -10.f), 10.f);
    if (!valid) logit = -INFINITY;

    float mx  = bmax64(logit, red);
    float ex  = expf(logit - mx);
    float ssm = bsum64(ex, red);
    float attn = ex / ssm;

    if (valid) atomicAdd(&weight[row], attn * (1.f / (float)NWIN));
}

// ---------------------------------------------------------------------------
// Fbar[l,e] = sum_p weight[l,p] * LN(feats[l*512+p])[e]
// blockIdx = l*8 + pchunk ; 128 threads x float4 (512B coalesced per wave).
// ---------------------------------------------------------------------------
__global__ void acc_kernel(const float* __restrict__ feats, const float* __restrict__ rmean,
                           const float* __restrict__ rrstd, const float* __restrict__ weight,
                           float* __restrict__ Fbar) {
    int bid = blockIdx.x;
    int l   = bid >> 3;
    int pc  = bid & 7;          // 8 chunks of 64 rows
    int e0  = threadIdx.x * 4;  // 128 threads cover all 512 dims

    float4 acc = make_float4(0.f, 0.f, 0.f, 0.f);
    int pbase = pc * 64;
    for (int pp = 0; pp < 64; ++pp) {
        int row = l * SEGLEN + pbase + pp;
        if (pp + 8 < 64)
            __builtin_prefetch(&feats[(size_t)(row + 8) * DDIM + e0], 0, 0);
        float a  = weight[row] * rrstd[row];
        float mb = rmean[row];
        float4 f = *(const float4*)(&feats[(size_t)row * DDIM + e0]);
        acc.x = fmaf(a, f.x - mb, acc.x);
        acc.y = fmaf(a, f.y - mb, acc.y);
        acc.z = fmaf(a, f.z - mb, acc.z);
        acc.w = fmaf(a, f.w - mb, acc.w);
    }
    float* dst = &Fbar[(size_t)l * DDIM + e0];
    atomicAdd(dst + 0, acc.x);
    atomicAdd(dst + 1, acc.y);
    atomicAdd(dst + 2, acc.z);
    atomicAdd(dst + 3, acc.w);
}

// ---------------------------------------------------------------------------
// Host launch
// ---------------------------------------------------------------------------
extern "C" void kernel_launch(void* const* d_in, const int* in_sizes, int n_in,
                              void* d_out, int out_size, void* d_ws, size_t ws_size,
                              hipStream_t stream) {
    (void)in_sizes; (void)n_in; (void)out_size; (void)ws_size;

    const float* feats  = (const float*)d_in[0];
    const float* coords = (const float*)d_in[1];
    // d_in[2] = mask (all-False, unused by reference math)
    const float* z   = (const float*)d_in[3];
    const float* Wq  = (const float*)d_in[4];
    const float* Wk  = (const float*)d_in[5];
    const float* Wv  = (const float*)d_in[6];
    const float* pw1 = (const float*)d_in[7];
    const float* pb1 = (const float*)d_in[8];
    const float* pw2 = (const float*)d_in[9];
    const float* pb2 = (const float*)d_in[10];
    const float* Wp  = (const float*)d_in[11];
    const float* bp  = (const float*)d_in[12];
    float* out = (float*)d_out;

    float* ws    = (float*)d_ws;
    float* cst   = ws;                    // 4
    float* q     = ws + 16;               // 64*512
    float* qk    = q    + LTOK * DDIM;
    float* qw2   = qk   + LTOK * DDIM;
    float* dotv  = qw2  + LTOK * DDIM;    // [64*512] per-row logit base
    float* wgt   = dotv + NROWS;          // [64*512] scattered attn weights
    float* Fbar  = wgt  + NROWS;          // [64*512]  (contiguous with wgt)
    float* Zb    = Fbar + LTOK * DDIM;
    float* rmean = Zb   + LTOK * DDIM;    // [32768]
    float* rrstd = rmean + NROWS;         // [32768]

    const int gtiles = (LTOK / 16) * (DDIM / 16);   // 128 tiles, 1 wave each

    coord_stats_kernel<<<1, 256, 0, stream>>>(coords, cst);
    zero_kernel<<<(2 * NROWS + 255) / 256, 256, 0, stream>>>(wgt, 2 * NROWS); // wgt + Fbar

    // q = z @ Wq^T ; qk = q @ Wk ; qw2 = q @ pos_w2
    gemm_wmma_f32<true,  false><<<gtiles, 32, 0, stream>>>(z,  Wq,  nullptr, q,   LTOK, DDIM, DDIM);
    gemm_wmma_f32<false, false><<<gtiles, 32, 0, stream>>>(q,  Wk,  nullptr, qk,  LTOK, DDIM, DDIM);
    gemm_wmma_f32<false, false><<<gtiles, 32, 0, stream>>>(q,  pw2, nullptr, qw2, LTOK, DDIM, DDIM);

    dot_kernel<<<NROWS / 8, 256, 0, stream>>>(feats, qk, dotv, rmean, rrstd);
    attn_kernel<<<LTOK * NWIN, 64, 0, stream>>>(coords, cst, q, qw2, pw1, pb1, pb2, dotv, wgt);
    acc_kernel<<<LTOK * 8, 128, 0, stream>>>(feats, rmean, rrstd, wgt, Fbar);

    // Z = Fbar @ Wv^T ; out = Z @ Wp^T + bp
    gemm_wmma_f32<true, false><<<gtiles, 32, 0, stream>>>(Fbar, Wv, nullptr, Zb,  LTOK, DDIM, DDIM);
    gemm_wmma_f32<true, true ><<<gtiles, 32, 0, stream>>>(Zb,  Wp, bp,      out, LTOK, DDIM, DDIM);
}